// yolov3layer_57114475102620
// MI455X (gfx1250) — compile-verified
//
#include <hip/hip_runtime.h>
#include <hip/hip_bf16.h>
#include <math.h>

// ---------------- problem constants (shapes fixed by setup_inputs) ----------
#define NBOX   22743        // 3*(19^2 + 38^2 + 76^2)
#define NB     16           // batch
#define NCLS   80
#define TOPK   100
#define L0END  1083         // 3*19*19
#define L1END  5415         // L0END + 3*38*38
#define INPSZ  608.0f       // 19*32

// workspace layout (float/int32 units)
#define OFF_BOXES 0u                     // NB*NBOX*4   = 1,455,552
#define OFF_TOPS  1455552u               // NB*80*100   =   128,000
#define OFF_TOPB  1583552u               // NB*80*100*4 =   512,000
#define OFF_CAND  2095552u               // NB*8000     =   128,000

typedef float v2f __attribute__((ext_vector_type(2)));
typedef float v8f __attribute__((ext_vector_type(8)));

#if defined(__HIP_DEVICE_COMPILE__) && __has_builtin(__builtin_amdgcn_wmma_f32_16x16x4_f32)
#define HAVE_WMMA4 1
#else
#define HAVE_WMMA4 0
#endif

// -------- helpers -----------------------------------------------------------
__device__ __forceinline__ float sigf(float x) { return 1.0f / (1.0f + expf(-x)); }

__device__ __forceinline__ void decode_n(int n, int& G, int& layer, int& a, int& gx, int& gy) {
    int off;
    if (n < L0END)      { G = 19; layer = 0; off = n; }
    else if (n < L1END) { G = 38; layer = 1; off = n - L0END; }
    else                { G = 76; layer = 2; off = n - L1END; }
    a = off % 3;
    int cell = off / 3;
    gx = cell % G;
    gy = cell / G;
}

__device__ __forceinline__ void transform_consts(const float* image_shape,
                                                 float& sy, float& sx, float& cy, float& cx) {
    float ih = image_shape[0], iw = image_shape[1];
    float r  = fminf(INPSZ / ih, INPSZ / iw);
    float nh = rintf(ih * r), nw = rintf(iw * r);
    float oy = (INPSZ - nh) * 0.5f / INPSZ;
    float ox = (INPSZ - nw) * 0.5f / INPSZ;
    sy = (INPSZ / nh) * ih;          // (box_y - oy)*scale_y * image_h
    sx = (INPSZ / nw) * iw;
    cy = -oy * sy;
    cx = -ox * sx;
}

// order-preserving float->u32 (total order, works for negatives/-inf)
__device__ __forceinline__ unsigned int ord32(float x) {
    unsigned int u = __float_as_uint(x);
    return (u & 0x80000000u) ? ~u : (u | 0x80000000u);
}
// key: larger score wins; ties -> lower index wins (matches jax.lax.top_k)
__device__ __forceinline__ unsigned long long packKey(float v, unsigned int idx) {
    return ((unsigned long long)ord32(v) << 32) | (unsigned long long)(0xFFFFFFFFu - idx);
}

// ---------------- kernel 1: decode boxes (WMMA affine transform) ------------
// One wave handles 16 boxes per V_WMMA_F32_16X16X4_F32:
//   D(4 coords x 16 boxes) = A(4x4 affine) x B(4 inputs x 16 boxes) + C(consts)
__global__ void decode_boxes_kernel(const float* __restrict__ f0,
                                    const float* __restrict__ f1,
                                    const float* __restrict__ f2,
                                    const float* __restrict__ anchors,
                                    const float* __restrict__ image_shape,
                                    float* __restrict__ boxes) {
    const int lane = threadIdx.x & 31;
    const int wave = threadIdx.x >> 5;
    const bool hi  = lane >= 16;
    const int  n16 = lane & 15;

    const long long total = (long long)NB * NBOX;
    long long g = ((long long)blockIdx.x * (blockDim.x >> 5) + wave) * 16 + n16;
    const bool valid = g < total;
    long long gc = valid ? g : 0;

    int b = (int)(gc / NBOX);
    int n = (int)(gc % NBOX);

    int G, layer, a, gx, gy;
    decode_n(n, G, layer, a, gx, gy);
    const float* f = (layer == 0) ? f0 : (layer == 1) ? f1 : f2;

    long long plane = (long long)G * G;
    long long base  = (((long long)b * 255 + a * 85) * G + gy) * G + gx;

    float tx = f[base];
    float ty = f[base + plane];
    float tw = f[base + 2 * plane];
    float th = f[base + 3 * plane];

    float xc = (sigf(tx) + (float)gx) / (float)G;
    float yc = (sigf(ty) + (float)gy) / (float)G;
    int aidx = (2 - layer) * 3 + a;
    float w = expf(tw) * anchors[aidx * 2 + 0] / INPSZ;
    float h = expf(th) * anchors[aidx * 2 + 1] / INPSZ;

    float sy, sx, cy, cx;
    transform_consts(image_shape, sy, sx, cy, cx);

#if HAVE_WMMA4
    // A (16x4, rows M, K split lanes0-15:K0/K1, lanes16-31:K2/K3)
    int m = lane & 15;
    float a0, a1;
    if (!hi) {   // K=0 (yc), K=1 (xc)
        a0 = (m == 0 || m == 2) ? sy : 0.0f;
        a1 = (m == 1 || m == 3) ? sx : 0.0f;
    } else {     // K=2 (h),  K=3 (w)
        a0 = (m == 0) ? -0.5f * sy : (m == 2) ? 0.5f * sy : 0.0f;
        a1 = (m == 1) ? -0.5f * sx : (m == 3) ? 0.5f * sx : 0.0f;
    }
    v2f A; A.x = a0; A.y = a1;
    // B (4x16): lane n supplies column n; low lanes rows K0/K1, high lanes K2/K3
    v2f Bm; Bm.x = hi ? h : yc; Bm.y = hi ? w : xc;
    // C: constant column broadcast into rows 0..3 (lanes 0-15 hold M=0..7)
    v8f C = {};
    if (!hi) { C[0] = cy; C[1] = cx; C[2] = cy; C[3] = cx; }
    v8f D = __builtin_amdgcn_wmma_f32_16x16x4_f32(false, A, false, Bm, (short)0, C,
                                                  false, false);
    if (!hi && valid) {
        float4 o; o.x = D[0]; o.y = D[1]; o.z = D[2]; o.w = D[3];
        *(float4*)(boxes + g * 4) = o;
    }
#else
    if (!hi && valid) {
        float4 o;
        o.x = sy * yc - 0.5f * sy * h + cy;   // y1
        o.y = sx * xc - 0.5f * sx * w + cx;   // x1
        o.z = sy * yc + 0.5f * sy * h + cy;   // y2
        o.w = sx * xc + 0.5f * sx * w + cx;   // x2
        *(float4*)(boxes + g * 4) = o;
    }
#endif
}

// ---------------- kernel 2: fused score + per-class top-100 -----------------
// One block per (image, class). Scores for all 22743 anchors cached in LDS
// (91 KB of the 320 KB/WGP), then 100 exact argmax sweeps via ds_max_u64.
__global__ void class_topk_kernel(const float* __restrict__ f0,
                                  const float* __restrict__ f1,
                                  const float* __restrict__ f2,
                                  const float* __restrict__ boxes,
                                  float* __restrict__ tops,
                                  float* __restrict__ topb) {
    extern __shared__ float s_scores[];          // NBOX floats (dynamic)
    __shared__ unsigned long long s_best;
    __shared__ int   s_topi[TOPK];
    __shared__ float s_topv[TOPK];

    const int bc = blockIdx.x;
    const int b  = bc / NCLS;
    const int c  = bc % NCLS;
    const int tid = threadIdx.x;
    const int nth = blockDim.x;

    // phase A: compute score = sigmoid(conf)*sigmoid(prob_c) for every anchor
    for (int n = tid; n < NBOX; n += nth) {
        int G, layer, a, gx, gy;
        decode_n(n, G, layer, a, gx, gy);
        const float* f = (layer == 0) ? f0 : (layer == 1) ? f1 : f2;
        long long plane = (long long)G * G;
        long long base  = (((long long)b * 255 + a * 85) * G + gy) * G + gx;
        float conf = f[base + 4 * plane];
        float prob = f[base + (long long)(5 + c) * plane];
        s_scores[n] = sigf(conf) * sigf(prob);
    }
    __syncthreads();

    // phase B: 100 exact argmax sweeps (tie-break: lowest index)
    for (int k = 0; k < TOPK; ++k) {
        if (tid == 0) s_best = 0ull;
        __syncthreads();
        unsigned long long loc = 0ull;
        for (int n = tid; n < NBOX; n += nth) {
            unsigned long long key = packKey(s_scores[n], (unsigned)n);
            if (key > loc) loc = key;
        }
        atomicMax(&s_best, loc);
        __syncthreads();
        if (tid == 0) {
            unsigned int idx = 0xFFFFFFFFu - (unsigned int)(s_best & 0xFFFFFFFFull);
            s_topi[k] = (int)idx;
            s_topv[k] = s_scores[idx];
            s_scores[idx] = -3.402823466e38f;    // exclude from later sweeps
        }
        __syncthreads();
    }

    // phase C: write scores + gather boxes of selected anchors
    for (int k = tid; k < TOPK; k += nth) {
        int idx = s_topi[k];
        tops[bc * TOPK + k] = s_topv[k];
        const float* src = boxes + ((long long)b * NBOX + idx) * 4;
        float* dst = topb + (size_t)(bc * TOPK + k) * 4;
        dst[0] = src[0]; dst[1] = src[1]; dst[2] = src[2]; dst[3] = src[3];
    }
}

// ---------------- kernel 3: greedy NMS per (image, class) -------------------
__global__ void nms_kernel(const float* __restrict__ tops,
                           const float* __restrict__ topb,
                           float* __restrict__ cand) {
    const int bc = blockIdx.x;
    __shared__ float y1s[TOPK], x1s[TOPK], y2s[TOPK], x2s[TOPK], ars[TOPK], scs[TOPK];
    __shared__ int sup[TOPK];

    const int j = threadIdx.x;
    if (j < TOPK) {
        const float* bb = topb + (size_t)(bc * TOPK + j) * 4;
        y1s[j] = bb[0]; x1s[j] = bb[1]; y2s[j] = bb[2]; x2s[j] = bb[3];
        ars[j] = fmaxf(y2s[j] - y1s[j], 0.0f) * fmaxf(x2s[j] - x1s[j], 0.0f);
        scs[j] = tops[bc * TOPK + j];
        sup[j] = 0;
    }
    __syncthreads();

    for (int i = 0; i < TOPK; ++i) {
        // sup[i] only ever written by thread i in earlier iterations (synced)
        bool keep_i = (scs[i] >= 0.2f) && (sup[i] == 0);
        if (j < TOPK && j > i && keep_i) {
            float iy1 = fmaxf(y1s[i], y1s[j]);
            float ix1 = fmaxf(x1s[i], x1s[j]);
            float iy2 = fminf(y2s[i], y2s[j]);
            float ix2 = fminf(x2s[i], x2s[j]);
            float inter = fmaxf(iy2 - iy1, 0.0f) * fmaxf(ix2 - ix1, 0.0f);
            float iou = inter / (ars[i] + ars[j] - inter + 1e-9f);
            if (iou > 0.3f) sup[j] = 1;          // only thread j writes sup[j]
        }
        __syncthreads();
    }

    if (j < TOPK) {
        bool keep = (scs[j] >= 0.2f) && (sup[j] == 0);
        cand[bc * TOPK + j] = keep ? scs[j] : -1.0f;
    }
}

// ---------------- kernel 4: final per-image top-100 -------------------------
// Candidates staged global->LDS with async-tensor path, then exact argmax.
__global__ void final_topk_kernel(const float* __restrict__ cand,
                                  const float* __restrict__ topb,
                                  float* __restrict__ out) {
    const int b = blockIdx.x;
    __shared__ __attribute__((aligned(16))) float s_cand[NCLS * TOPK];   // 32 KB
    __shared__ unsigned long long s_best;
    const int tid = threadIdx.x;
    const int nth = blockDim.x;

    const float* src = cand + (size_t)b * (NCLS * TOPK);
    // async copy: 2000 x b128 (16B per lane) global -> LDS
    for (int q = tid; q < (NCLS * TOPK) / 4; q += nth) {
        unsigned int lds_addr = (unsigned int)(unsigned long long)(size_t)&s_cand[q * 4];
        unsigned long long gaddr = (unsigned long long)(size_t)(src + q * 4);
        asm volatile("global_load_async_to_lds_b128 %0, %1, off"
                     :: "v"(lds_addr), "v"(gaddr) : "memory");
    }
    asm volatile("s_wait_asynccnt 0" ::: "memory");
    __syncthreads();

    for (int k = 0; k < TOPK; ++k) {
        if (tid == 0) s_best = 0ull;
        __syncthreads();
        unsigned long long loc = 0ull;
        for (int q = tid; q < NCLS * TOPK; q += nth) {
            unsigned long long key = packKey(s_cand[q], (unsigned)q);
            if (key > loc) loc = key;
        }
        atomicMax(&s_best, loc);
        __syncthreads();
        if (tid == 0) {
            unsigned int idx = 0xFFFFFFFFu - (unsigned int)(s_best & 0xFFFFFFFFull);
            float sv = s_cand[idx];
            const float* bb = topb + ((size_t)b * (NCLS * TOPK) + idx) * 4;
            float* o = out + (size_t)(b * TOPK + k) * 6;
            o[0] = bb[0]; o[1] = bb[1]; o[2] = bb[2]; o[3] = bb[3];
            o[4] = sv;
            o[5] = (float)(idx / TOPK);
            s_cand[idx] = -3.402823466e38f;
        }
        __syncthreads();
    }
}

// ---------------- host launcher ---------------------------------------------
extern "C" void kernel_launch(void* const* d_in, const int* in_sizes, int n_in,
                              void* d_out, int out_size, void* d_ws, size_t ws_size,
                              hipStream_t stream) {
    const float* f0       = (const float*)d_in[0];   // (16,255,19,19)
    const float* f1       = (const float*)d_in[1];   // (16,255,38,38)
    const float* f2       = (const float*)d_in[2];   // (16,255,76,76)
    const float* anchors  = (const float*)d_in[3];   // (9,2)
    const float* imshape  = (const float*)d_in[4];   // (2,)
    float* out = (float*)d_out;                      // (16,100,6)

    float* ws    = (float*)d_ws;
    float* boxes = ws + OFF_BOXES;
    float* tops  = ws + OFF_TOPS;
    float* topb  = ws + OFF_TOPB;
    float* cand  = ws + OFF_CAND;

    // k1: 22743 waves of 16 boxes; 8 waves/block
    int waves = (NB * NBOX + 15) / 16;               // 22743
    int blocks1 = (waves + 7) / 8;                   // 2843
    decode_boxes_kernel<<<blocks1, 256, 0, stream>>>(f0, f1, f2, anchors, imshape, boxes);

    // k2: one block per (image, class); 91 KB dynamic LDS score cache
    size_t shmem = (size_t)NBOX * sizeof(float);     // 90,972 B
    class_topk_kernel<<<NB * NCLS, 256, shmem, stream>>>(f0, f1, f2, boxes, tops, topb);

    // k3: NMS
    nms_kernel<<<NB * NCLS, 128, 0, stream>>>(tops, topb, cand);

    // k4: final selection
    final_topk_kernel<<<NB, 256, 0, stream>>>(cand, topb, out);
}